// MinkUNet_4063039062848
// MI455X (gfx1250) — compile-verified
//
#include <hip/hip_runtime.h>

// ---------------------------------------------------------------------------
// Types for CDNA5 WMMA fragments
// ---------------------------------------------------------------------------
typedef float v2f __attribute__((ext_vector_type(2)));
typedef float v8f __attribute__((ext_vector_type(8)));

#define CI_CHUNK 64  // channels of B staged in LDS per step

// ---------------------------------------------------------------------------
// Generic sparse conv:  out[Nout,Co] = sum_k gather(x, nbr[:,k]) @ W[k]
//   x   : [Nin, Ci]   f32
//   W   : [K, Ci, Co] f32
//   nbr : [Nout, K]   int32, -1 = missing neighbor; nullptr => identity (K==1)
// One wave => 16 output rows x (NT*16) output cols via V_WMMA_F32_16X16X4_F32.
// Block = 256 threads = 8 waves = 8 consecutive row tiles sharing one column
// group; the W[k] slice is staged in LDS once per block and reused 8x.
// LDS layout interleaves each even/odd K-row pair so a B fragment is one
// aligned ds_load_b64:  lds[pair][2*col + {0,1}] = W[2p][col], W[2p+1][col].
// All index math is 32-bit (every tensor here is < 2^31 elements).
// ---------------------------------------------------------------------------
template <int NT>
__global__ __launch_bounds__(256) void sconv_kernel(
    const float* __restrict__ x, const float* __restrict__ W,
    const int* __restrict__ nbr, float* __restrict__ out,
    int Nout, int Ci, int Co, int K)
{
#if defined(__gfx1250__) && __has_builtin(__builtin_amdgcn_wmma_f32_16x16x4_f32)
  constexpr int TW   = NT * 16;       // tile width in output channels
  constexpr int STR2 = 2 * TW + 4;    // LDS stride per interleaved K-row pair
  __shared__ float ldsB[(CI_CHUNK / 2) * STR2];

  const int tid     = threadIdx.x;
  const int lane    = tid & 31;
  const int l15     = lane & 15;
  const int half    = lane >> 4;                 // 0 or 1 (K interleave)
  const int rowTile = blockIdx.x * 8 + (tid >> 5);
  const int colBase = blockIdx.y * TW;
  const int row     = rowTile * 16 + l15;        // A-fragment row (M = lane%16)

  v8f acc[NT] = {};

  int idxNext = (row < Nout) ? (nbr ? nbr[row * K] : row) : -1;

  for (int k = 0; k < K; ++k) {
    const int idx = idxNext;
    if (k + 1 < K) {
      idxNext = (row < Nout) ? (nbr ? nbr[row * K + k + 1] : row) : -1;
      if (idxNext >= 0)
        __builtin_prefetch(x + idxNext * Ci, 0, 1);  // global_prefetch
    }
    const float amask = (idx >= 0) ? 1.0f : 0.0f;    // zero row for -1
    const float* xr   = x + (idx < 0 ? 0 : idx) * Ci;
    const float* Wk   = W + k * Ci * Co;

    for (int ci0 = 0; ci0 < Ci; ci0 += CI_CHUNK) {
      const int chunk = (Ci - ci0 < CI_CHUNK) ? (Ci - ci0) : CI_CHUNK;
      __syncthreads();
      // cooperative stage of B tile [chunk][TW], K-row pairs interleaved,
      // zero-padded past Co
      for (int e = tid; e < chunk * TW; e += 256) {
        const int cc = e / TW, col = e % TW;
        const int gcol = colBase + col;
        ldsB[(cc >> 1) * STR2 + 2 * col + (cc & 1)] =
            (gcol < Co) ? Wk[(ci0 + cc) * Co + gcol] : 0.0f;
      }
      __syncthreads();

      for (int cc = 0; cc < chunk; cc += 4) {
        // A 16x4 f32 fragment: lane holds row M=l15, K = {2*half, 2*half+1}
        v2f a = *(const v2f*)(xr + ci0 + cc + 2 * half) * amask;
        const int q = (cc >> 1) + half;  // interleaved pair index for this lane
#pragma unroll
        for (int t = 0; t < NT; ++t) {
          // B 4x16 fragment: one b64 from the interleaved pair slot
          v2f b = *(const v2f*)&ldsB[q * STR2 + 2 * (t * 16 + l15)];
          acc[t] = __builtin_amdgcn_wmma_f32_16x16x4_f32(
              false, a, false, b, (short)0, acc[t], false, false);
        }
      }
    }
  }

  // store: C/D layout -> vgpr j, lane(l15,half) holds (M = j + 8*half, N = l15)
  const int mBase = rowTile * 16 + 8 * half;
  if (rowTile * 16 + 16 <= Nout && colBase + TW <= Co) {
    // fast path: full tile, no guards, one base address
    float* op = out + mBase * Co + colBase + l15;
#pragma unroll
    for (int t = 0; t < NT; ++t)
#pragma unroll
      for (int j = 0; j < 8; ++j) op[t * 16 + j * Co] = acc[t][j];
  } else {
#pragma unroll
    for (int t = 0; t < NT; ++t) {
      const int col = colBase + t * 16 + l15;
      if (col >= Co) continue;
#pragma unroll
      for (int j = 0; j < 8; ++j) {
        const int m = mBase + j;
        if (m < Nout) out[m * Co + col] = acc[t][j];
      }
    }
  }
#else
  // scalar fallback (also what the host pass parses) -- same output coverage
  const int l15f     = threadIdx.x & 15;
  const int halff    = (threadIdx.x >> 4) & 1;
  const int rowTileF = blockIdx.x * 8 + (threadIdx.x >> 5);
  for (int t = 0; t < NT; ++t) {
    const int col = blockIdx.y * (NT * 16) + t * 16 + l15f;
    if (col >= Co) continue;
    for (int j = 0; j < 8; ++j) {
      const int m = rowTileF * 16 + j + 8 * halff;
      if (m >= Nout) continue;
      float s = 0.f;
      for (int k = 0; k < K; ++k) {
        const int idx = nbr ? nbr[m * K + k] : m;
        if (idx < 0) continue;
        const float* xr = x + idx * Ci;
        const float* wk = W + k * Ci * Co + col;
        for (int ci = 0; ci < Ci; ++ci) s += xr[ci] * wk[ci * Co];
      }
      out[m * Co + col] = s;
    }
  }
#endif
}

// ---------------------------------------------------------------------------
// Deterministic two-stage BatchNorm statistics (training mode, eps=1e-5).
// Stage 1: 256 fixed row-slabs -> partial [slab][2C] (sum, sumsq)
// Stage 2: single block reduces slabs in fixed order, emits fused scale/shift:
//          y = x*scale[c] + shift[c],  scale=g*rsqrt(var+eps), shift=b-mu*scale
// ---------------------------------------------------------------------------
__global__ void bn_partial_kernel(const float* __restrict__ t, int N, int C,
                                  float* __restrict__ partial, int slabs)
{
  const int slab = blockIdx.x;
  const int r0 = (int)((long long)slab * N / slabs);
  const int r1 = (int)((long long)(slab + 1) * N / slabs);
  for (int c = threadIdx.x; c < C; c += blockDim.x) {
    float s = 0.f, q = 0.f;
    const float* p = t + r0 * C + c;
    for (int r = r0; r < r1; ++r, p += C) {
      const float v = *p;
      s += v;
      q += v * v;
    }
    partial[slab * 2 * C + c]     = s;
    partial[slab * 2 * C + C + c] = q;
  }
}

__global__ void bn_finalize_kernel(const float* __restrict__ partial, int slabs,
                                   int N, int C, const float* __restrict__ g,
                                   const float* __restrict__ b,
                                   float* __restrict__ ss)
{
  for (int c = threadIdx.x; c < C; c += blockDim.x) {
    float s = 0.f, q = 0.f;
    for (int sl = 0; sl < slabs; ++sl) {
      s += partial[sl * 2 * C + c];
      q += partial[sl * 2 * C + C + c];
    }
    const float inv = 1.0f / (float)N;
    const float mu  = s * inv;
    const float var = q * inv - mu * mu;
    const float sc  = g[c] * rsqrtf(var + 1e-5f);
    ss[c]     = sc;
    ss[C + c] = b[c] - mu * sc;
  }
}

// ---------------------------------------------------------------------------
// Fused elementwise kernels: one row per block, threads sweep channels.
// (no runtime div/mod, all 32-bit addressing)
// ---------------------------------------------------------------------------
__global__ void bn_apply_kernel(const float* __restrict__ t,
                                const float* __restrict__ ss,
                                float* __restrict__ dst, int C, int Cd,
                                int coff, int relu)
{
  const int r = blockIdx.x;
  const float* s = t + r * C;
  float* d = dst + r * Cd + coff;
  for (int c = threadIdx.x; c < C; c += blockDim.x) {
    float v = s[c] * ss[c] + ss[C + c];
    if (relu) v = fmaxf(v, 0.f);
    d[c] = v;
  }
}

__global__ void bn_add2_kernel(const float* __restrict__ a,
                               const float* __restrict__ ssa,
                               const float* __restrict__ b,
                               const float* __restrict__ ssb,
                               float* __restrict__ out, int C)
{
  const int r = blockIdx.x;
  const float* pa = a + r * C;
  const float* pb = b + r * C;
  float* po = out + r * C;
  for (int c = threadIdx.x; c < C; c += blockDim.x) {
    const float va = pa[c] * ssa[c] + ssa[C + c];
    const float vb = pb[c] * ssb[c] + ssb[C + c];
    po[c] = fmaxf(va + vb, 0.f);
  }
}

__global__ void bn_addskip_kernel(const float* __restrict__ a,
                                  const float* __restrict__ ssa,
                                  const float* __restrict__ skip,
                                  float* __restrict__ out, int C)
{
  const int r = blockIdx.x;
  const float* pa = a + r * C;
  const float* pk = skip + r * C;
  float* po = out + r * C;
  for (int c = threadIdx.x; c < C; c += blockDim.x)
    po[c] = fmaxf(pa[c] * ssa[c] + ssa[C + c] + pk[c], 0.f);
}

__global__ void copy_strided_kernel(const float* __restrict__ src,
                                    float* __restrict__ dst, int C, int Cd,
                                    int coff)
{
  const int r = blockIdx.x;
  const float* s = src + r * C;
  float* d = dst + r * Cd + coff;
  for (int c = threadIdx.x; c < C; c += blockDim.x) d[c] = s[c];
}

// ---------------------------------------------------------------------------
// Host-side orchestration
// ---------------------------------------------------------------------------
namespace {

struct P3 { const float *w, *g, *b; };

struct Net {
  hipStream_t st;
  float *partial, *ss0, *ss1;
};

inline int ebs(int C) { return C <= 64 ? 64 : (C <= 128 ? 128 : 256); }

void sconv(Net& n, const float* x, const float* W, const int* nbr, float* out,
           int Nout, int Ci, int Co, int K)
{
  const int gb = (Nout + 127) / 128;
  if (Co % 128 == 0) {
    dim3 g(gb, Co / 128);
    sconv_kernel<8><<<g, 256, 0, n.st>>>(x, W, nbr, out, Nout, Ci, Co, K);
  } else if (Co % 64 == 0) {
    dim3 g(gb, Co / 64);
    sconv_kernel<4><<<g, 256, 0, n.st>>>(x, W, nbr, out, Nout, Ci, Co, K);
  } else {
    dim3 g(gb, (Co + 31) / 32);
    sconv_kernel<2><<<g, 256, 0, n.st>>>(x, W, nbr, out, Nout, Ci, Co, K);
  }
}

void bnstats(Net& n, const float* t, int N, int C, const float* g,
             const float* b, float* ss)
{
  bn_partial_kernel<<<256, 256, 0, n.st>>>(t, N, C, n.partial, 256);
  bn_finalize_kernel<<<1, 384, 0, n.st>>>(n.partial, 256, N, C, g, b, ss);
}

void bnapply(Net& n, const float* t, const float* ss, float* dst, int N, int C,
             int Cd, int coff, int relu)
{
  bn_apply_kernel<<<N, ebs(C), 0, n.st>>>(t, ss, dst, C, Cd, coff, relu);
}

}  // namespace

extern "C" void kernel_launch(void* const* d_in, const int* in_sizes, int n_in,
                              void* d_out, int out_size, void* d_ws,
                              size_t ws_size, hipStream_t stream)
{
  (void)n_in; (void)out_size; (void)ws_size;
  auto F = [&](int i) { return (const float*)d_in[i]; };
  auto I = [&](int i) { return (const int*)d_in[i]; };

  // ---- walk params in construction order -------------------------------
  int pi = 1;  // d_in[0] == x
  auto cb = [&]() { P3 r{F(pi), F(pi + 1), F(pi + 2)}; pi += 3; return r; };

  P3 stem = cb();
  P3 down[4], r1c1[4], r1c2[4], r1ds[4] = {}, r2c1[4], r2c2[4];
  const int eci[4] = {32, 32, 64, 128};
  const int ec1[4] = {32, 64, 128, 256};
  for (int s = 0; s < 4; ++s) {
    down[s] = cb();
    r1c1[s] = cb();
    r1c2[s] = cb();
    if (eci[s] != ec1[s]) r1ds[s] = cb();
    r2c1[s] = cb();
    r2c2[s] = cb();
  }
  P3 ude[4], ur1c1[4], ur1c2[4], ur1ds[4], ur2c1[4], ur2c2[4];
  for (int u = 0; u < 4; ++u) {
    ude[u]   = cb();
    ur1c1[u] = cb();
    ur1c2[u] = cb();
    ur1ds[u] = cb();  // decoder r1 always has downsample (ccat != co)
    ur2c1[u] = cb();
    ur2c2[u] = cb();
  }
  const float* clsw = F(pi); pi += 1;

  // ---- maps -------------------------------------------------------------
  const int* mc5 = I(pi); pi += 1;
  const int* mc3[5];
  int idx_c3[5];
  for (int l = 0; l < 5; ++l) { idx_c3[l] = pi; mc3[l] = I(pi); pi += 1; }
  const int *mdown[4], *mup[4];
  for (int l = 0; l < 4; ++l) { mdown[l] = I(pi); pi += 1; mup[l] = I(pi); pi += 1; }

  int N[5];
  N[0] = in_sizes[0] / 4;
  for (int l = 1; l < 5; ++l) N[l] = in_sizes[idx_c3[l]] / 27;

  // ---- workspace layout (bump allocator, deterministic) -----------------
  char* wsBase = (char*)d_ws;
  size_t wsOff = 0;
  auto alloc = [&](long long elems) -> float* {
    float* p = (float*)(wsBase + wsOff);
    wsOff += ((size_t)elems * 4 + 255) & ~(size_t)255;
    return p;
  };

  Net net;
  net.st      = stream;
  net.partial = alloc(256LL * 2 * 384);
  net.ss0     = alloc(2 * 384);
  net.ss1     = alloc(2 * 384);

  float* enc0 = alloc((long long)N[0] * 32);
  float* enc1 = alloc((long long)N[1] * 32);
  float* enc2 = alloc((long long)N[2] * 64);
  float* enc3 = alloc((long long)N[3] * 128);
  float* enc4 = alloc((long long)N[4] * 256);

  long long maxT = 0;
  auto mx = [&](long long v) { if (v > maxT) maxT = v; };
  mx((long long)N[0] * 128); mx((long long)N[1] * 128);
  mx((long long)N[2] * 192); mx((long long)N[3] * 384);
  mx((long long)N[4] * 256);
  float* t0 = alloc(maxT);
  float* t1 = alloc(maxT);
  float* t2 = alloc(maxT);
  float* t3 = alloc(maxT);

  // ---- composite helpers ------------------------------------------------
  auto cbr = [&](const float* in, P3 p, const int* nbr, int K, int Nn, int Ci,
                 int Co, float* buf) {
    sconv(net, in, p.w, nbr, buf, Nn, Ci, Co, K);
    bnstats(net, buf, Nn, Co, p.g, p.b, net.ss0);
    bnapply(net, buf, net.ss0, buf, Nn, Co, Co, 0, 1);
  };

  auto res = [&](const float* in, int Nn, int Ci, int Co, const int* nbr, P3 c1,
                 P3 c2, const P3* ds, float* tA, float* tB, float* out) {
    sconv(net, in, c1.w, nbr, tA, Nn, Ci, Co, 27);
    bnstats(net, tA, Nn, Co, c1.g, c1.b, net.ss0);
    bnapply(net, tA, net.ss0, tA, Nn, Co, Co, 0, 1);
    sconv(net, tA, c2.w, nbr, tB, Nn, Co, Co, 27);
    bnstats(net, tB, Nn, Co, c2.g, c2.b, net.ss0);
    if (ds) {
      sconv(net, in, ds->w, nullptr, tA, Nn, Ci, Co, 1);  // 1x1 identity map
      bnstats(net, tA, Nn, Co, ds->g, ds->b, net.ss1);
      bn_add2_kernel<<<Nn, ebs(Co), 0, stream>>>(tB, net.ss0, tA, net.ss1, out,
                                                 Co);
    } else {
      bn_addskip_kernel<<<Nn, ebs(Co), 0, stream>>>(tB, net.ss0, in, out, Co);
    }
  };

  auto decat = [&](const float* hin, P3 de, const int* up, int Nf, int Ci,
                   int Co, const float* enc, int Cenc, float* tmp, float* cat) {
    const int Ccat = Co + Cenc;
    sconv(net, hin, de.w, up, tmp, Nf, Ci, Co, 8);
    bnstats(net, tmp, Nf, Co, de.g, de.b, net.ss0);
    bnapply(net, tmp, net.ss0, cat, Nf, Co, Ccat, 0, 1);
    copy_strided_kernel<<<Nf, ebs(Cenc), 0, stream>>>(enc, cat, Cenc, Ccat, Co);
  };

  // ---- forward ----------------------------------------------------------
  const float* x = F(0);

  // stem (k=5^3, 4->32) directly into skip buffer enc0
  cbr(x, stem, mc5, 125, N[0], 4, 32, enc0);

  // encoder
  cbr(enc0, down[0], mdown[0], 8, N[1], 32, 32, t0);
  res(t0, N[1], 32, 32, mc3[1], r1c1[0], r1c2[0], nullptr, t2, t3, t1);
  res(t1, N[1], 32, 32, mc3[1], r2c1[0], r2c2[0], nullptr, t2, t3, enc1);

  cbr(enc1, down[1], mdown[1], 8, N[2], 32, 32, t0);
  res(t0, N[2], 32, 64, mc3[2], r1c1[1], r1c2[1], &r1ds[1], t2, t3, t1);
  res(t1, N[2], 64, 64, mc3[2], r2c1[1], r2c2[1], nullptr, t2, t3, enc2);

  cbr(enc2, down[2], mdown[2], 8, N[3], 64, 64, t0);
  res(t0, N[3], 64, 128, mc3[3], r1c1[2], r1c2[2], &r1ds[2], t2, t3, t1);
  res(t1, N[3], 128, 128, mc3[3], r2c1[2], r2c2[2], nullptr, t2, t3, enc3);

  cbr(enc3, down[3], mdown[3], 8, N[4], 128, 128, t0);
  res(t0, N[4], 128, 256, mc3[4], r1c1[3], r1c2[3], &r1ds[3], t2, t3, t1);
  res(t1, N[4], 256, 256, mc3[4], r2c1[3], r2c2[3], nullptr, t2, t3, enc4);

  // decoder
  decat(enc4, ude[0], mup[3], N[3], 256, 256, enc3, 128, t0, t1);
  res(t1, N[3], 384, 256, mc3[3], ur1c1[0], ur1c2[0], &ur1ds[0], t2, t3, t0);
  res(t0, N[3], 256, 256, mc3[3], ur2c1[0], ur2c2[0], nullptr, t2, t3, t1);

  decat(t1, ude[1], mup[2], N[2], 256, 128, enc2, 64, t0, t2);
  res(t2, N[2], 192, 128, mc3[2], ur1c1[1], ur1c2[1], &ur1ds[1], t0, t3, t1);
  res(t1, N[2], 128, 128, mc3[2], ur2c1[1], ur2c2[1], nullptr, t0, t3, t2);

  decat(t2, ude[2], mup[1], N[1], 128, 96, enc1, 32, t0, t1);
  res(t1, N[1], 128, 96, mc3[1], ur1c1[2], ur1c2[2], &ur1ds[2], t0, t3, t2);
  res(t2, N[1], 96, 96, mc3[1], ur2c1[2], ur2c2[2], nullptr, t0, t3, t1);

  decat(t1, ude[3], mup[0], N[0], 96, 96, enc0, 32, t0, t2);
  res(t2, N[0], 128, 96, mc3[0], ur1c1[3], ur1c2[3], &ur1ds[3], t0, t3, t1);
  res(t1, N[0], 96, 96, mc3[0], ur2c1[3], ur2c2[3], nullptr, t0, t3, t2);

  // classifier: [N0,96] @ [96,20]
  sconv(net, t2, clsw, nullptr, (float*)d_out, N[0], 96, 20, 1);
}